// NCCLoss_19164144075053
// MI455X (gfx1250) — compile-verified
//
#include <hip/hip_runtime.h>
#include <stdint.h>

typedef __attribute__((ext_vector_type(2))) float v2f;
typedef __attribute__((ext_vector_type(8))) float v8f;

#define WIN   9
#define PAD   4
#define OT    8            // output tile edge
#define IT    16           // input tile edge = OT + 2*PAD
#define NN    192
#define NB    (NN / OT)    // 24 blocks per dimension
#define INV_WIN_SIZE (1.0f / 729.0f)

__device__ __forceinline__ v8f wmma4(v2f a, v2f b, v8f c) {
    return __builtin_amdgcn_wmma_f32_16x16x4_f32(
        false, a, false, b, (short)0, c, false, false);
}

// LDS pool (18432 floats = 72 KB), regions aliased across pipeline phases:
//   phase 1 (load + x-filter):  sI [0,4096)  sJ [4096,8192)  stA5 [8192,18432)
//   phase 2 (y-filter):         stB5 [0,5120)            reads stA5
//   phase 3 (z-filter):         out5 [5120,7680)         reads stB5
//   phase 4 (ncc reduce):       red  [7680,7936)         reads out5
__global__ __launch_bounds__(256) void ncc_tile_kernel(
    const float* __restrict__ I, const float* __restrict__ J,
    float* __restrict__ partial)
{
    __shared__ float pool[18432];
    float* const sIp  = pool;                 // 4096 floats
    float* const sJp  = pool + 4096;          // 4096 floats
    float* const stA  = pool + 8192;          // 5*2048: [c][z][y][xo]
    float* const stB  = pool;                 // 5*1024: [c][z][yo][xo]
    float* const outc = pool + 5120;          // 5*512:  [c][zo*64+yo*8+xo]
    float* const red  = pool + 7680;          // 256

    const int tid = threadIdx.x;
    const int bx = blockIdx.x, by = blockIdx.y, bz = blockIdx.z;
    const int x0 = bx * OT - PAD, y0 = by * OT - PAD, z0 = bz * OT - PAD;

    const bool interior = (bx >= 1) && (bx <= NB - 2) &&
                          (by >= 1) && (by <= NB - 2) &&
                          (bz >= 1) && (bz <= NB - 2);

    if (interior) {
        // CDNA5 async copy global->LDS, 16B per lane-op, GVS addressing.
        const uint64_t ibase = (uint64_t)I;
        const uint64_t jbase = (uint64_t)J;
        for (int cid = tid; cid < (IT * IT * IT) / 4; cid += 256) {
            const int z = cid >> 6, rem = cid & 63, y = rem >> 2, xc = rem & 3;
            const uint32_t goff =
                (uint32_t)((((z0 + z) * NN + (y0 + y)) * NN + x0 + xc * 4) * 4);
            const uint32_t la = (uint32_t)(uintptr_t)&sIp[cid * 4];
            const uint32_t lb = (uint32_t)(uintptr_t)&sJp[cid * 4];
            asm volatile("global_load_async_to_lds_b128 %0, %1, %2"
                         :: "v"(la), "v"(goff), "s"(ibase) : "memory");
            asm volatile("global_load_async_to_lds_b128 %0, %1, %2"
                         :: "v"(lb), "v"(goff), "s"(jbase) : "memory");
        }
        asm volatile("s_wait_asynccnt 0" ::: "memory");
    } else {
        // boundary tiles: guarded loads, zero padding (matches SAME conv)
        for (int i = tid; i < IT * IT * IT; i += 256) {
            const int z = i >> 8, rem = i & 255, y = rem >> 4, x = rem & 15;
            const int gz = z0 + z, gy = y0 + y, gx = x0 + x;
            const bool inb = (gx >= 0) && (gx < NN) && (gy >= 0) && (gy < NN) &&
                             (gz >= 0) && (gz < NN);
            const long gidx = ((long)gz * NN + gy) * NN + gx;
            sIp[i] = inb ? I[gidx] : 0.0f;
            sJp[i] = inb ? J[gidx] : 0.0f;
        }
    }
    __syncthreads();

    const int lane = tid & 31;
    const int wave = tid >> 5;       // 8 waves
    const int half = lane >> 4;      // 0: lanes 0-15, 1: lanes 16-31
    const int lm   = lane & 15;      // A row (M) / B,D column (N)

    // ===== x-filter as banded GEMM, all 5 channels per z-slice =============
    // Out(16y x 16xo) = P(16y x 16x) * W(16x x 16xo), W[x][xo]=1 iff xo<=x<=xo+8
    // A 16x4 f32 layout: VGPR0 K=k+2*half, VGPR1 K=k+2*half+1, row M=lane%16.
    #pragma unroll
    for (int zi = 0; zi < 2; ++zi) {
        const int z = wave + zi * 8;
        // Load this lane's 8 needed row elements once: pairs at k=4p+2*half.
        const float2* rI =
            (const float2*)(sIp + (z * IT + lm) * IT + 2 * half);
        const float2* rJ =
            (const float2*)(sJp + (z * IT + lm) * IT + 2 * half);
        float ai[8], aj[8], bw[8];
        #pragma unroll
        for (int p = 0; p < 4; ++p) {
            const float2 vi = rI[2 * p];
            const float2 vj = rJ[2 * p];
            ai[2 * p] = vi.x; ai[2 * p + 1] = vi.y;
            aj[2 * p] = vj.x; aj[2 * p + 1] = vj.y;
            const int kk = 4 * p + 2 * half;
            bw[2 * p]     = (lm <= kk     && kk     <= lm + 8) ? 1.0f : 0.0f;
            bw[2 * p + 1] = (lm <= kk + 1 && kk + 1 <= lm + 8) ? 1.0f : 0.0f;
        }
        v8f acc0 = {}, acc1 = {}, acc2 = {}, acc3 = {}, acc4 = {};
        #pragma unroll
        for (int p = 0; p < 4; ++p) {
            const float i0 = ai[2 * p], i1 = ai[2 * p + 1];
            const float j0 = aj[2 * p], j1 = aj[2 * p + 1];
            v2f b; b.x = bw[2 * p]; b.y = bw[2 * p + 1];
            v2f a;
            a.x = i0;      a.y = i1;      acc0 = wmma4(a, b, acc0);  // I
            a.x = j0;      a.y = j1;      acc1 = wmma4(a, b, acc1);  // J
            a.x = i0 * i0; a.y = i1 * i1; acc2 = wmma4(a, b, acc2);  // I*I
            a.x = j0 * j0; a.y = j1 * j1; acc3 = wmma4(a, b, acc3);  // J*J
            a.x = i0 * j0; a.y = i1 * j1; acc4 = wmma4(a, b, acc4);  // I*J
        }
        // D layout: VGPR r, lane L -> (M = r + 8*(L>=16), N = L%16)
        if (lm < OT) {
            #pragma unroll
            for (int r = 0; r < 8; ++r) {
                const int y = r + half * 8;
                const int o = (z * IT + y) * OT + lm;
                stA[0 * 2048 + o] = acc0[r];
                stA[1 * 2048 + o] = acc1[r];
                stA[2 * 2048 + o] = acc2[r];
                stA[3 * 2048 + o] = acc3[r];
                stA[4 * 2048 + o] = acc4[r];
            }
        }
    }
    __syncthreads();

    // ===== y-filter (9-tap sums), all channels ==============================
    for (int i = tid; i < 5 * IT * OT * OT; i += 256) {
        const int c = i >> 10, r = i & 1023;
        const int z = r >> 6, rem = r & 63, yo = rem >> 3, xo = rem & 7;
        float s = 0.0f;
        #pragma unroll
        for (int k = 0; k < WIN; ++k)
            s += stA[((c * IT + z) * IT + yo + k) * OT + xo];
        stB[i] = s;
    }
    __syncthreads();

    // ===== z-filter =========================================================
    for (int i = tid; i < 5 * OT * OT * OT; i += 256) {
        const int c = i >> 9, r = i & 511;
        const int zo = r >> 6, rem = r & 63;
        float s = 0.0f;
        #pragma unroll
        for (int k = 0; k < WIN; ++k)
            s += stB[c * 1024 + (zo + k) * 64 + rem];
        outc[i] = s;
    }
    __syncthreads();

    // ===== per-voxel NCC + block reduction ==================================
    float lsum = 0.0f;
    for (int i = tid; i < OT * OT * OT; i += 256) {
        const float Is = outc[0 * 512 + i], Js = outc[1 * 512 + i];
        const float I2 = outc[2 * 512 + i], J2 = outc[3 * 512 + i];
        const float IJ = outc[4 * 512 + i];
        const float cross = IJ - Is * Js * INV_WIN_SIZE;
        const float iv    = I2 - Is * Is * INV_WIN_SIZE;
        const float jv    = J2 - Js * Js * INV_WIN_SIZE;
        lsum += cross * cross / (iv * jv + 1e-5f);
    }
    red[tid] = lsum;
    __syncthreads();
    for (int s = 128; s > 0; s >>= 1) {
        if (tid < s) red[tid] += red[tid + s];
        __syncthreads();
    }
    if (tid == 0)
        partial[(bz * NB + by) * NB + bx] = red[0];
}

__global__ __launch_bounds__(256) void ncc_reduce_kernel(
    const float* __restrict__ partial, float* __restrict__ out, int n)
{
    __shared__ float red[256];
    float s = 0.0f;
    for (int i = threadIdx.x; i < n; i += 256) s += partial[i];
    red[threadIdx.x] = s;
    __syncthreads();
    for (int k = 128; k > 0; k >>= 1) {
        if (threadIdx.x < k) red[threadIdx.x] += red[threadIdx.x + k];
        __syncthreads();
    }
    if (threadIdx.x == 0)
        out[0] = -red[0] / (float)((long)NN * NN * NN);
}

extern "C" void kernel_launch(void* const* d_in, const int* in_sizes, int n_in,
                              void* d_out, int out_size, void* d_ws, size_t ws_size,
                              hipStream_t stream) {
    (void)in_sizes; (void)n_in; (void)out_size; (void)ws_size;
    const float* I = (const float*)d_in[0];   // y_true
    const float* J = (const float*)d_in[1];   // y_pred
    float* out     = (float*)d_out;
    float* partial = (float*)d_ws;            // NB^3 floats = 54 KB scratch

    dim3 grid(NB, NB, NB);
    ncc_tile_kernel<<<grid, 256, 0, stream>>>(I, J, partial);
    ncc_reduce_kernel<<<1, 256, 0, stream>>>(partial, out, NB * NB * NB);
}